// VectorQuantizer_48266842472527
// MI455X (gfx1250) — compile-verified
//
#include <hip/hip_runtime.h>

typedef float v2f __attribute__((ext_vector_type(2)));
typedef float v8f __attribute__((ext_vector_type(8)));

#define NROWS 262144
#define DIM   64
#define KCODES 512
#define ROWS_PER_WAVE 16
#define WAVES_PER_BLOCK 8
#define ROWS_PER_BLOCK (ROWS_PER_WAVE * WAVES_PER_BLOCK)   // 128
#define NBLOCKS (NROWS / ROWS_PER_BLOCK)                   // 2048

// Fused VQ kernel: per-wave 16-row tile, f32 WMMA dot products against all 512
// codes (two independent accumulator chains for ILP), in-register argmin,
// gather + straight-through output, deterministic loss partials.
__global__ __launch_bounds__(256) void vq_main_kernel(
    const float* __restrict__ x,    // [N, 64]
    const float* __restrict__ E,    // [64, 512] row-major
    float* __restrict__ out,        // [N, 64]
    float* __restrict__ partial)    // [NBLOCKS] loss partial sums
{
  __shared__ float s_e2[KCODES];
  __shared__ int   s_idx[WAVES_PER_BLOCK][ROWS_PER_WAVE];
  __shared__ float s_loss[WAVES_PER_BLOCK];

  const int tid  = threadIdx.x;
  const int wave = tid >> 5;
  const int lane = tid & 31;
  const int half = lane >> 4;   // 0: lanes 0-15, 1: lanes 16-31
  const int l16  = lane & 15;

  // ---- Phase 0: e2[k] = sum_d E[d,k]^2 (coalesced, block-cooperative) ----
  for (int c = tid; c < KCODES; c += 256) {
    float acc = 0.0f;
#pragma unroll
    for (int d = 0; d < DIM; ++d) {
      float v = E[d * KCODES + c];
      acc += v * v;
    }
    s_e2[c] = acc;
  }
  __syncthreads();

  const int rowBase = blockIdx.x * ROWS_PER_BLOCK + wave * ROWS_PER_WAVE;

  // ---- Phase 1: load A fragments for V_WMMA_F32_16X16X4_F32 ----
  // A layout (16x4 f32): lanes 0-15 row M=lane hold K={0,1}; lanes 16-31 row
  // M=lane-16 hold K={2,3}.  For k-step kk the k-base is 4*kk.
  v2f a[16];
  {
    const float* xp = x + (size_t)(rowBase + l16) * DIM + 2 * half;
#pragma unroll
    for (int kk = 0; kk < 16; ++kk)
      a[kk] = *(const v2f*)(xp + 4 * kk);
  }

  float bmin[8];
  int   bidx[8];
#pragma unroll
  for (int j = 0; j < 8; ++j) { bmin[j] = 3.4e38f; bidx[j] = 0; }

  // ---- Phase 2: 32 column tiles of 16 codes, processed 2 at a time ----
  // B layout (4x16 f32): lanes 0-15 col N=lane hold K={0,1}; lanes 16-31 col
  // N=lane-16 hold K={2,3}.  E is [D,K] row-major, so element (kr, col) is at
  // E[kr*512 + col]; the two per-lane values are 2048B apart (strided b32).
  // Two tiles -> two independent WMMA accumulator chains (hides WMMA latency).
  const float* ebase = E + 2 * half * KCODES + l16;
  for (int t = 0; t < 32; t += 2) {
    const int col0 = t * 16 + l16;
    const int col1 = col0 + 16;
    const float* eb0 = ebase + t * 16;
    const float* eb1 = eb0 + 16;
    v8f c0 = {0.f, 0.f, 0.f, 0.f, 0.f, 0.f, 0.f, 0.f};
    v8f c1 = {0.f, 0.f, 0.f, 0.f, 0.f, 0.f, 0.f, 0.f};
#pragma unroll
    for (int kk = 0; kk < 16; ++kk) {
      v2f b0, b1;
      b0.x = eb0[(4 * kk + 0) * KCODES];
      b0.y = eb0[(4 * kk + 1) * KCODES];
      b1.x = eb1[(4 * kk + 0) * KCODES];
      b1.y = eb1[(4 * kk + 1) * KCODES];
      c0 = __builtin_amdgcn_wmma_f32_16x16x4_f32(
          false, a[kk], false, b0, (short)0, c0, false, false);
      c1 = __builtin_amdgcn_wmma_f32_16x16x4_f32(
          false, a[kk], false, b1, (short)0, c1, false, false);
    }
    // C layout: VGPR j holds row (j + 8*half), column colX, for this lane.
    const float e2c0 = s_e2[col0];
    const float e2c1 = s_e2[col1];
#pragma unroll
    for (int j = 0; j < 8; ++j) {
      float s0 = e2c0 - 2.0f * c0[j];
      if (s0 < bmin[j]) { bmin[j] = s0; bidx[j] = col0; }
      float s1 = e2c1 - 2.0f * c1[j];
      if (s1 < bmin[j]) { bmin[j] = s1; bidx[j] = col1; }
    }
  }

  // ---- Phase 3: argmin reduction across the 16 lanes of each half ----
  // xor masks 1,2,4,8 never cross bit4, so halves stay independent.
#pragma unroll
  for (int j = 0; j < 8; ++j) {
#pragma unroll
    for (int off = 1; off < 16; off <<= 1) {
      float ov = __shfl_xor(bmin[j], off, 32);
      int   oi = __shfl_xor(bidx[j], off, 32);
      if (ov < bmin[j] || (ov == bmin[j] && oi < bidx[j])) {
        bmin[j] = ov; bidx[j] = oi;
      }
    }
  }
  if (l16 == 0) {
#pragma unroll
    for (int j = 0; j < 8; ++j) s_idx[wave][8 * half + j] = bidx[j];
  }
  __syncthreads();

  // ---- Phase 4: gather codebook rows, write straight-through output, loss ----
  float lacc = 0.0f;
  const int d0 = lane * 2;
  for (int j = 0; j < ROWS_PER_WAVE; ++j) {
    const int r = rowBase + j;
    const int k = s_idx[wave][j];
    float q0 = E[(d0 + 0) * KCODES + k];
    float q1 = E[(d0 + 1) * KCODES + k];
    v2f xv = *(const v2f*)(x + (size_t)r * DIM + d0);
    float dq0 = q0 - xv.x;
    float dq1 = q1 - xv.y;
    v2f o;
    o.x = xv.x + dq0;   // x + stop_gradient(q - x), reference rounding
    o.y = xv.y + dq1;
    *(v2f*)(out + (size_t)r * DIM + d0) = o;
    lacc += dq0 * dq0 + dq1 * dq1;
  }

  // wave-level sum, then block-level sum (deterministic order)
#pragma unroll
  for (int off = 1; off < 32; off <<= 1) lacc += __shfl_xor(lacc, off, 32);
  if (lane == 0) s_loss[wave] = lacc;
  __syncthreads();
  if (tid == 0) {
    float s = 0.0f;
#pragma unroll
    for (int w = 0; w < WAVES_PER_BLOCK; ++w) s += s_loss[w];
    partial[blockIdx.x] = s;
  }
}

// Deterministic final reduction of the 2048 per-block partials -> loss scalar.
__global__ __launch_bounds__(256) void vq_loss_reduce_kernel(
    const float* __restrict__ partial, float* __restrict__ loss)
{
  __shared__ float s[8];
  float acc = 0.0f;
  for (int i = threadIdx.x; i < NBLOCKS; i += 256) acc += partial[i];
#pragma unroll
  for (int off = 1; off < 32; off <<= 1) acc += __shfl_xor(acc, off, 32);
  const int wave = threadIdx.x >> 5;
  const int lane = threadIdx.x & 31;
  if (lane == 0) s[wave] = acc;
  __syncthreads();
  if (threadIdx.x == 0) {
    float t = 0.0f;
#pragma unroll
    for (int w = 0; w < 8; ++w) t += s[w];
    // loss = 0.25*commitment + codebook, and both equal mean((q-x)^2):
    loss[0] = t * (1.25f / ((float)NROWS * (float)DIM));
  }
}

extern "C" void kernel_launch(void* const* d_in, const int* in_sizes, int n_in,
                              void* d_out, int out_size, void* d_ws, size_t ws_size,
                              hipStream_t stream) {
  const float* x = (const float*)d_in[0];       // [262144, 64] f32
  const float* E = (const float*)d_in[1];       // [64, 512]  f32
  float* out     = (float*)d_out;               // [262144*64] quantized_st + [1] loss
  float* partial = (float*)d_ws;                // NBLOCKS floats of scratch

  vq_main_kernel<<<NBLOCKS, 256, 0, stream>>>(x, E, out, partial);
  vq_loss_reduce_kernel<<<1, 256, 0, stream>>>(partial, out + (size_t)NROWS * DIM);
}